// PredLoss_5875515260973
// MI455X (gfx1250) — compile-verified
//
#include <hip/hip_runtime.h>
#include <hip/hip_bf16.h>
#include <math.h>

#define NO_LANE 90
#define M_ 6
#define T_ 30
#define L_ 10
#define P_ 50
#define K2_WAVES 8

typedef __attribute__((ext_vector_type(2))) float v2f;
typedef __attribute__((ext_vector_type(8))) float v8f;

// ds_swizzle group-of-32: offset = (xor<<10) | and(0x1f), or=0  -> src = lane ^ xor
template<int IMM> __device__ __forceinline__ float swzf(float x) {
  return __int_as_float(__builtin_amdgcn_ds_swizzle(__float_as_int(x), IMM));
}
template<int IMM> __device__ __forceinline__ int swzi(int x) {
  return __builtin_amdgcn_ds_swizzle(x, IMM);
}

__global__ void __launch_bounds__(32) init_out(float* out, float* ws) {
  int i = threadIdx.x;
  if (i < 8) out[i] = 0.f;
  if (i < 2) ws[i] = 0.f;
}

// One thread per batch element: cls_loss num/den, reg_loss, num_reg, lane_cls_loss, num_lane_cls
__global__ void __launch_bounds__(256) loss_main(
    const float* __restrict__ cls, const float* __restrict__ reg,
    const float* __restrict__ lane_cls, const float* __restrict__ gt,
    const unsigned char* __restrict__ has, const int* __restrict__ lab,
    float* __restrict__ out, float* __restrict__ ws, int B)
{
  int b = blockIdx.x * 256 + threadIdx.x;
  float c_num = 0.f, c_den = 0.f, r_loss = 0.f, n_reg = 0.f, lc_loss = 0.f, n_lc = 0.f;
  if (b < B) {
    const float2* reg2 = (const float2*)reg;
    const float2* gt2  = (const float2*)gt;
    float gx[T_], gy[T_], hf[T_];
    int last = 0; float hsum = 0.f;
#pragma unroll
    for (int t = 0; t < T_; t++) {
      float2 g = gt2[b * T_ + t];
      gx[t] = g.x; gy[t] = g.y;
      float hv = (float)has[b * T_ + t];
      hf[t] = hv; hsum += hv;
      if (hv > 0.5f) last = t;            // argmax of hasf + 0.1*t/T
    }
    float maskf = (last > 0) ? 1.f : 0.f; // has[:,0] always true
    float2 eg = gt2[b * T_ + last];
    int tar = 0, minm = 0;
    float bestS = 3.4e38f, bestE = 3.4e38f;
#pragma unroll
    for (int m = 0; m < M_; m++) {
      const float2* rrow = reg2 + ((size_t)b * M_ + m) * T_;
      float s = 0.f;
#pragma unroll
      for (int t = 0; t < T_; t++) {
        float2 r = rrow[t];
        float dx = r.x - gx[t], dy = r.y - gy[t];
        s += dx * dx + dy * dy;
      }
      if (s < bestS) { bestS = s; tar = m; }
      float2 er = rrow[last];
      float ex = er.x - eg.x, ey = er.y - eg.y;
      float ed = sqrtf(ex * ex + ey * ey);
      if (ed < bestE) { bestE = ed; minm = m; }
    }
    // -log_softmax(cls)[tar]
    float cv[M_]; float mx = -3.4e38f;
#pragma unroll
    for (int m = 0; m < M_; m++) { cv[m] = cls[b * M_ + m]; mx = fmaxf(mx, cv[m]); }
    float se = 0.f;
#pragma unroll
    for (int m = 0; m < M_; m++) se += expf(cv[m] - mx);
    float lse = mx + logf(se);
    float ctar = cls[b * M_ + tar];
    c_num = (lse - ctar) * maskf;
    c_den = maskf;
    // smooth-L1 regression loss on selected mode
    const float2* rsel = reg2 + ((size_t)b * M_ + minm) * T_;
#pragma unroll
    for (int t = 0; t < T_; t++) {
      float2 r = rsel[t];
      float w = hf[t] * maskf;
      float dx = r.x - gx[t], dy = r.y - gy[t];
      float ax = fabsf(dx), ay = fabsf(dy);
      float lx = (ax < 1.f) ? 0.5f * dx * dx : (ax - 0.5f);
      float ly = (ay < 1.f) ? 0.5f * dy * dy : (ay - 0.5f);
      r_loss += (lx + ly) * w;
    }
    n_reg = hsum * maskf;
    // lane classification loss
    int lv = lab[b];
    bool valid = (lv != NO_LANE);
    int labc = valid ? lv : 0;
    float lcv[L_]; float lmx = -3.4e38f;
#pragma unroll
    for (int i = 0; i < L_; i++) { lcv[i] = lane_cls[b * L_ + i]; lmx = fmaxf(lmx, lcv[i]); }
    float lse2 = 0.f;
#pragma unroll
    for (int i = 0; i < L_; i++) lse2 += expf(lcv[i] - lmx);
    float lnll = lmx + logf(lse2) - lane_cls[b * L_ + labc];
    if (valid) { lc_loss = lnll * ((float)labc + 1.f); n_lc = 1.f; }
  }
  __shared__ float s6[6];
  if (threadIdx.x < 6) s6[threadIdx.x] = 0.f;
  __syncthreads();
  atomicAdd(&s6[0], c_num); atomicAdd(&s6[1], c_den);
  atomicAdd(&s6[2], r_loss); atomicAdd(&s6[3], n_reg);
  atomicAdd(&s6[4], lc_loss); atomicAdd(&s6[5], n_lc);
  __syncthreads();
  if (threadIdx.x == 0) {
    atomicAdd(&ws[0], s6[0]); atomicAdd(&ws[1], s6[1]);
    atomicAdd(&out[2], s6[2]); atomicAdd(&out[3], s6[3]);
    atomicAdd(&out[4], s6[4]); atomicAdd(&out[5], s6[5]);
  }
}

// One wave per batch element. Cross terms X·Y^T via V_WMMA_F32_16X16X4_F32:
// X rows: [reg_sel t=0..29 | pad | gt t=0..29 | pad] (64), Y rows: lane_w p=0..49 | pad (64).
// A layout: lanes 0-15 = rows M, VGPR0/1 = K0/K1; lanes 16-31 = K2/K3 (zero-padded).
// B layout: lanes 0-15 = cols N with K0/K1 in VGPR0/1; lanes 16-31 = K2/K3 (zero).
__global__ void __launch_bounds__(256) lane_off_kernel(
    const float* __restrict__ reg, const float* __restrict__ gt,
    const float* __restrict__ rot, const float* __restrict__ orig,
    const float* __restrict__ lane_feats, const unsigned char* __restrict__ has,
    const int* __restrict__ lab, float* __restrict__ ws_per, float* __restrict__ ws_w, int B)
{
  int wid = threadIdx.x >> 5;
  int lane = threadIdx.x & 31;
  int b = blockIdx.x * K2_WAVES + wid;
  int L16 = lane & 15;
  float per = 0.f, woff = 0.f;
  if (b < B) {
    int hv = (lane < T_) ? (int)has[b * T_ + lane] : 0;
    int lastc = hv ? lane : 0;
    { int s;
      s = swzi<0x041f>(lastc); lastc = lastc > s ? lastc : s;
      s = swzi<0x081f>(lastc); lastc = lastc > s ? lastc : s;
      s = swzi<0x101f>(lastc); lastc = lastc > s ? lastc : s;
      s = swzi<0x201f>(lastc); lastc = lastc > s ? lastc : s;
      s = swzi<0x401f>(lastc); lastc = lastc > s ? lastc : s; }
    int last = lastc;
    int lv = lab[b];
    bool active = (last > 0) && (lv != NO_LANE);   // wave-uniform
    if (active) {
      const float2* reg2 = (const float2*)reg;
      const float2* gt2  = (const float2*)gt;
      const float2* lf2  = (const float2*)lane_feats;
      float2 eg = gt2[b * T_ + last];
      int minm = 0; float bestE = 3.4e38f;
#pragma unroll
      for (int m = 0; m < M_; m++) {
        float2 er = reg2[((size_t)b * M_ + m) * T_ + last];
        float ex = er.x - eg.x, ey = er.y - eg.y;
        float ed = sqrtf(ex * ex + ey * ey);
        if (ed < bestE) { bestE = ed; minm = m; }
      }
      float r00 = rot[b * 4 + 0], r01 = rot[b * 4 + 1];
      float r10 = rot[b * 4 + 2], r11 = rot[b * 4 + 3];
      float o0 = orig[b * 2 + 0], o1 = orig[b * 2 + 1];
      // rotated lane points for this lane's column p = 16j + L16 (all lanes compute norms)
      float y0a[4], y1a[4], y2a[4];
#pragma unroll
      for (int j = 0; j < 4; j++) {
        int p = 16 * j + L16;
        if (p < P_) {
          float2 ls = lf2[((size_t)b * L_ + lv) * P_ + p];
          float y0 = fmaf(ls.x, r00, fmaf(ls.y, r10, o0));
          float y1 = fmaf(ls.x, r01, fmaf(ls.y, r11, o1));
          y0a[j] = y0; y1a[j] = y1; y2a[j] = y0 * y0 + y1 * y1;
        } else { y0a[j] = 0.f; y1a[j] = 0.f; y2a[j] = 3.0e38f; }
      }
      // A fragments (K=2 real, lanes>=16 feed zero K=2,3)
      float a0v[4], a1v[4];
#pragma unroll
      for (int i = 0; i < 4; i++) {
        float a0 = 0.f, a1 = 0.f;
        if (lane < 16) {
          int m = 16 * i + lane;
          if (m < T_)                      { float2 r = reg2[((size_t)b * M_ + minm) * T_ + m]; a0 = r.x; a1 = r.y; }
          else if (m >= 32 && m < 32 + T_) { float2 g = gt2[b * T_ + (m - 32)];                 a0 = g.x; a1 = g.y; }
        }
        a0v[i] = a0; a1v[i] = a1;
      }
      float rm[4][8];
#pragma unroll
      for (int i = 0; i < 4; i++)
#pragma unroll
        for (int r = 0; r < 8; r++) rm[i][r] = 3.0e38f;
      bool lo = (lane < 16);
#pragma unroll
      for (int i = 0; i < 4; i++) {
        v2f A; A.x = a0v[i]; A.y = a1v[i];
#pragma unroll
        for (int j = 0; j < 4; j++) {
          v2f Bf; Bf.x = lo ? y0a[j] : 0.f; Bf.y = lo ? y1a[j] : 0.f;
          v8f C = {0.f, 0.f, 0.f, 0.f, 0.f, 0.f, 0.f, 0.f};
          C = __builtin_amdgcn_wmma_f32_16x16x4_f32(false, A, false, Bf, (short)0, C, false, false);
          float yy = y2a[j];
#pragma unroll
          for (int r = 0; r < 8; r++)
            rm[i][r] = fminf(rm[i][r], yy - 2.f * C[r]);   // |y|^2 - 2 x·y (col = this lane's p)
        }
      }
      // min over columns: butterfly within each 16-lane half
#pragma unroll
      for (int i = 0; i < 4; i++)
#pragma unroll
        for (int r = 0; r < 8; r++) {
          float v = rm[i][r];
          v = fminf(v, swzf<0x041f>(v));
          v = fminf(v, swzf<0x081f>(v));
          v = fminf(v, swzf<0x101f>(v));
          v = fminf(v, swzf<0x201f>(v));
          rm[i][r] = v;
        }
      int half = lane >> 4;   // C/D: lanes 0-15 hold rows r, lanes 16-31 rows r+8
      float acc = 0.f, hs = 0.f;
#pragma unroll
      for (int i = 0; i < 2; i++)
#pragma unroll
        for (int r = 0; r < 8; r++) {
          int t = 16 * i + 8 * half + r;
          if (t < T_) {
            float2 xr = reg2[((size_t)b * M_ + minm) * T_ + t];
            float dr2 = fmaxf(xr.x * xr.x + xr.y * xr.y + rm[i][r], 0.f);
            float2 xg = gt2[b * T_ + t];
            float dg2 = fmaxf(xg.x * xg.x + xg.y * xg.y + rm[i + 2][r], 0.f);
            float dr = sqrtf(dr2), dg = sqrtf(dg2);
            float hvt = (float)has[b * T_ + t];
            acc += ((dr >= dg) ? (dr - dg) : 0.f) * hvt;
            hs += hvt;
          }
        }
      acc += swzf<0x401f>(acc);  // fold halves (disjoint t sets)
      hs  += swzf<0x401f>(hs);
      per = acc / fmaxf(hs, 1.f);
      woff = 1.f;
    }
  }
  __shared__ float sp[K2_WAVES], sw[K2_WAVES];
  if (lane == 0) { sp[wid] = per; sw[wid] = woff; }
  __syncthreads();
  if (threadIdx.x == 0) {
    float tp = 0.f, tw = 0.f;
#pragma unroll
    for (int w = 0; w < K2_WAVES; w++) { tp += sp[w]; tw += sw[w]; }
    ws_per[blockIdx.x] = tp; ws_w[blockIdx.x] = tw;
  }
}

__global__ void __launch_bounds__(256) finalize(
    const float* __restrict__ ws, const float* __restrict__ ws_per,
    const float* __restrict__ ws_w, float* __restrict__ out, int nblk)
{
  __shared__ float sp, sw;
  if (threadIdx.x == 0) { sp = 0.f; sw = 0.f; }
  __syncthreads();
  float tp = 0.f, tw = 0.f;
  for (int i = threadIdx.x; i < nblk; i += 256) { tp += ws_per[i]; tw += ws_w[i]; }
  atomicAdd(&sp, tp); atomicAdd(&sw, tw);
  __syncthreads();
  if (threadIdx.x == 0) {
    out[6] = sp;
    out[7] = sw;
    out[0] = ws[0] / fmaxf(ws[1], 1.f);
    out[1] = 1.f;
  }
}

extern "C" void kernel_launch(void* const* d_in, const int* in_sizes, int n_in,
                              void* d_out, int out_size, void* d_ws, size_t ws_size,
                              hipStream_t stream) {
  (void)n_in; (void)out_size; (void)ws_size;
  const float* cls        = (const float*)d_in[0];
  const float* reg        = (const float*)d_in[1];
  const float* lane_cls   = (const float*)d_in[2];
  const float* gt         = (const float*)d_in[3];
  const float* rot        = (const float*)d_in[4];
  const float* orig       = (const float*)d_in[5];
  const float* lane_feats = (const float*)d_in[6];
  const unsigned char* has = (const unsigned char*)d_in[7];
  const int* lab          = (const int*)d_in[8];
  float* out = (float*)d_out;
  float* ws  = (float*)d_ws;
  int B = in_sizes[0] / M_;
  int nblk2 = (B + K2_WAVES - 1) / K2_WAVES;
  float* ws_per = ws + 8;
  float* ws_w   = ws_per + nblk2;

  hipLaunchKernelGGL(init_out, dim3(1), dim3(32), 0, stream, out, ws);
  hipLaunchKernelGGL(loss_main, dim3((B + 255) / 256), dim3(256), 0, stream,
                     cls, reg, lane_cls, gt, has, lab, out, ws, B);
  hipLaunchKernelGGL(lane_off_kernel, dim3(nblk2), dim3(256), 0, stream,
                     reg, gt, rot, orig, lane_feats, has, lab, ws_per, ws_w, B);
  hipLaunchKernelGGL(finalize, dim3(1), dim3(256), 0, stream, ws, ws_per, ws_w, out, nblk2);
}